// SSD_50311246905829
// MI455X (gfx1250) — compile-verified
//
#include <hip/hip_runtime.h>
#include <math.h>

// SSD300 post-processing for MI455X (gfx1250, wave32).
// Latency/bandwidth-bound workload (no matmul structure -> WMMA not applicable);
// CDNA5 paths used: global_load_async_to_lds_b128 (ASYNCcnt), s_wait_asynccnt,
// wave32 shuffle reductions, large LDS staging.

#define BATCH   64
#define NBOX    8732
#define NCLS    21
#define TOPK    200
#define BS      256
#define NWAVE   (BS / 32)
#define CONF_TH 0.01f
#define NMS_TH  0.45f

// Monotone float -> unsigned mapping (preserves order incl. -inf).
__device__ __forceinline__ unsigned ford(float f) {
    unsigned u = __float_as_uint(f);
    return (u & 0x80000000u) ? ~u : (u | 0x80000000u);
}

// Raw LDS byte offset of a __shared__ object (what async-to-LDS wants in VDST).
typedef __attribute__((address_space(3))) void lds_void_t;
__device__ __forceinline__ unsigned lds_byte_off(void* p) {
    return (unsigned)(unsigned long long)(lds_void_t*)p;
}

// Selection key: high 32 = order-preserved score bits, low 32 = ~index so the
// max key is (max score, lowest index) -- identical tie order to jax.lax.top_k.
__device__ __forceinline__ unsigned long long skey(float s, int n) {
    return ((unsigned long long)ford(s) << 32) |
           (unsigned long long)(0xFFFFFFFFu - (unsigned)n);
}

// ---------------------------------------------------------------------------
// Kernel A: softmax once per (b, n); store *masked* class scores class-major
// [B, 20, NBOX] so each NMS block can DMA a contiguous row into LDS.
// ---------------------------------------------------------------------------
__global__ void __launch_bounds__(BS)
ssd_scores_kernel(const float* __restrict__ conf, float* __restrict__ sc) {
    int g = blockIdx.x * BS + threadIdx.x;
    if (g >= BATCH * NBOX) return;
    int b = g / NBOX;
    int n = g - b * NBOX;
    const float* cr = conf + ((size_t)b * NBOX + n) * NCLS;

    float x[NCLS];
    float m = -INFINITY;
#pragma unroll
    for (int c = 0; c < NCLS; ++c) { x[c] = cr[c]; m = fmaxf(m, x[c]); }
    float s = 0.f;
#pragma unroll
    for (int c = 0; c < NCLS; ++c) { x[c] = expf(x[c] - m); s += x[c]; }
#pragma unroll
    for (int c = 1; c < NCLS; ++c) {
        float p = x[c] / s;  // softmax prob (matches reference e/sum)
        sc[((size_t)b * (NCLS - 1) + (c - 1)) * NBOX + n] =
            (p > CONF_TH) ? p : -INFINITY;   // pre-masked like reference
    }
}

// ---------------------------------------------------------------------------
// Kernel B: per (batch, class) top-200 + greedy NMS + stable compaction.
// grid = BATCH * NCLS blocks; class 0 is the zero background row.
// ---------------------------------------------------------------------------
template <bool USE_SC>
__global__ void __launch_bounds__(BS)
ssd_nms_kernel(const float* __restrict__ loc, const float* __restrict__ conf,
               const float* __restrict__ dbox, const float* __restrict__ sc,
               float* __restrict__ out) {
    __shared__ __align__(16) float ls[NBOX];      // masked scores (35 KB)
    __shared__ int                 s_topi[TOPK];
    __shared__ float               s_tops[TOPK];
    __shared__ float               s_bx[TOPK][4];
    __shared__ float               s_area[TOPK];
    __shared__ int                 s_keep[TOPK];
    __shared__ int                 s_rank[TOPK];
    __shared__ unsigned long long  s_red[NWAVE];
    __shared__ int                 s_bidx;
    __shared__ float               s_orow[TOPK * 5];

    const int tid = threadIdx.x;
    const int bc  = blockIdx.x;
    const int b   = bc / NCLS;
    const int cls = bc - b * NCLS;          // output class index 0..20

    float* orow = out + (size_t)bc * TOPK * 5;

    if (cls == 0) {                         // background row: zeros
        for (int i = tid; i < TOPK * 5; i += BS) orow[i] = 0.f;
        return;
    }

    // ---- Phase 1: masked score row -> LDS ---------------------------------
    if (USE_SC) {
        // CDNA5 async DMA path: row is 16B-aligned (NBOX*4 = 2183*16).
        const float* srow = sc + ((size_t)b * (NCLS - 1) + (cls - 1)) * NBOX;
        const int N4 = NBOX / 4;  // 2183 float4 chunks
        for (int i = tid; i < N4; i += BS) {
            unsigned lo = lds_byte_off(&ls[i * 4]);
            const float* gp = srow + i * 4;
            asm volatile("global_load_async_to_lds_b128 %0, %1, off"
                         :: "v"(lo), "v"(gp) : "memory");
        }
        asm volatile("s_wait_asynccnt 0" ::: "memory");
    } else {
        // Fallback: recompute softmax for this class in-block.
        for (int n = tid; n < NBOX; n += BS) {
            const float* cr = conf + ((size_t)b * NBOX + n) * NCLS;
            float x[NCLS];
            float m = -INFINITY;
#pragma unroll
            for (int c = 0; c < NCLS; ++c) { x[c] = cr[c]; m = fmaxf(m, x[c]); }
            float s = 0.f, xc = 0.f;
#pragma unroll
            for (int c = 0; c < NCLS; ++c) {
                float e = expf(x[c] - m);
                s += e;
                if (c == cls) xc = e;
            }
            float p = xc / s;
            ls[n] = (p > CONF_TH) ? p : -INFINITY;
        }
    }
    __syncthreads();

    // ---- Phase 2: top-200 selection via cached stripe maxima --------------
    // Each thread owns the strided stripe n = tid, tid+BS, ... and caches its
    // best key. After an extraction only the owning stripe is stale, so only
    // that one thread rescans (35 elements); all other cached maxima survive.
    unsigned long long myBest = 0ull;
    for (int n = tid; n < NBOX; n += BS)
        if (skey(ls[n], n) > myBest) myBest = skey(ls[n], n);

    for (int k = 0; k < TOPK; ++k) {
        unsigned long long best = myBest;
#pragma unroll
        for (int off = 16; off > 0; off >>= 1) {   // wave32 reduce
            unsigned long long o = __shfl_xor(best, off, 32);
            if (o > best) best = o;
        }
        if ((tid & 31) == 0) s_red[tid >> 5] = best;
        __syncthreads();
        if (tid == 0) {
            unsigned long long bk = s_red[0];
#pragma unroll
            for (int w = 1; w < NWAVE; ++w)
                if (s_red[w] > bk) bk = s_red[w];
            int idx = (int)(0xFFFFFFFFu - (unsigned)(bk & 0xFFFFFFFFull));
            s_topi[k] = idx;
            s_tops[k] = ls[idx];
            ls[idx]   = -INFINITY;   // remove from further rounds
            s_bidx    = idx;
        }
        __syncthreads();
        int idx = s_bidx;
        if ((idx & (BS - 1)) == tid) {       // owner refreshes its stripe max
            myBest = 0ull;
            for (int n = tid; n < NBOX; n += BS)
                if (skey(ls[n], n) > myBest) myBest = skey(ls[n], n);
        }
        // Owner's ls reads race-free: bounded by the barrier above (publishes
        // the -inf write) and next iteration's barrier (before thread 0's next
        // ls write).
    }

    // ---- Phase 3: decode only the 200 selected boxes ----------------------
    float mx1 = 0.f, my1 = 0.f, mx2 = 0.f, my2 = 0.f, mar = 0.f;
    if (tid < TOPK) {
        int idx = s_topi[tid];
        const float4 l = ((const float4*)loc)[(size_t)b * NBOX + idx];
        const float4 d = ((const float4*)dbox)[idx];
        float cx = d.x + 0.1f * l.x * d.z;
        float cy = d.y + 0.1f * l.y * d.w;
        float w  = d.z * expf(0.2f * l.z);
        float h  = d.w * expf(0.2f * l.w);
        mx1 = cx - w * 0.5f;
        my1 = cy - h * 0.5f;
        mx2 = mx1 + w;
        my2 = my1 + h;
        mar = (mx2 - mx1) * (my2 - my1);
        s_bx[tid][0] = mx1; s_bx[tid][1] = my1;
        s_bx[tid][2] = mx2; s_bx[tid][3] = my2;
        s_area[tid]  = mar;
        s_keep[tid]  = 1;
    }
    __syncthreads();

    // ---- Phase 4: greedy NMS (IoU computed on the fly) --------------------
    for (int j = 0; j < TOPK; ++j) {
        bool kj = (s_keep[j] != 0) && (s_tops[j] > CONF_TH);
        if (kj && tid < TOPK && tid > j) {
            float xx1 = fmaxf(s_bx[j][0], mx1);
            float yy1 = fmaxf(s_bx[j][1], my1);
            float xx2 = fminf(s_bx[j][2], mx2);
            float yy2 = fminf(s_bx[j][3], my2);
            float inter = fmaxf(xx2 - xx1, 0.f) * fmaxf(yy2 - yy1, 0.f);
            float uni   = s_area[j] + mar - inter;
            if (inter / uni > NMS_TH) s_keep[tid] = 0;  // NaN compares false, as in jnp
        }
        __syncthreads();
    }

    // ---- Phase 5: stable compaction + row store ---------------------------
    if (tid < TOPK)
        s_keep[tid] = (s_keep[tid] && (s_tops[tid] > CONF_TH)) ? 1 : 0;
    __syncthreads();
    if (tid == 0) {
        int nk = 0;
        for (int i = 0; i < TOPK; ++i) s_rank[i] = s_keep[i] ? nk++ : -1;
    }
    __syncthreads();
    for (int i = tid; i < TOPK * 5; i += BS) s_orow[i] = 0.f;
    __syncthreads();
    if (tid < TOPK && s_keep[tid]) {
        int r = s_rank[tid];
        s_orow[r * 5 + 0] = s_tops[tid];
        s_orow[r * 5 + 1] = mx1;
        s_orow[r * 5 + 2] = my1;
        s_orow[r * 5 + 3] = mx2;
        s_orow[r * 5 + 4] = my2;
    }
    __syncthreads();
    for (int i = tid; i < TOPK * 5; i += BS) orow[i] = s_orow[i];
}

// ---------------------------------------------------------------------------
extern "C" void kernel_launch(void* const* d_in, const int* in_sizes, int n_in,
                              void* d_out, int out_size, void* d_ws, size_t ws_size,
                              hipStream_t stream) {
    (void)in_sizes; (void)n_in; (void)out_size;
    const float* loc  = (const float*)d_in[0];   // [64, 8732, 4]
    const float* conf = (const float*)d_in[1];   // [64, 8732, 21]
    const float* dbox = (const float*)d_in[2];   // [8732, 4]
    float* out = (float*)d_out;                  // [64, 21, 200, 5]

    const size_t sc_bytes = (size_t)BATCH * (NCLS - 1) * NBOX * sizeof(float);
    dim3 gridB(BATCH * NCLS), block(BS);

    if (ws_size >= sc_bytes) {
        float* sc = (float*)d_ws;
        dim3 gridA((BATCH * NBOX + BS - 1) / BS);
        ssd_scores_kernel<<<gridA, block, 0, stream>>>(conf, sc);
        ssd_nms_kernel<true><<<gridB, block, 0, stream>>>(loc, conf, dbox, sc, out);
    } else {
        ssd_nms_kernel<false><<<gridB, block, 0, stream>>>(loc, conf, dbox, nullptr, out);
    }
}